// MultihotEmbedding_16458314678493
// MI455X (gfx1250) — compile-verified
//
#include <hip/hip_runtime.h>

// MultihotEmbedding / per-row token histogram for MI455X (gfx1250).
//
// out[b, v] = #{ l : x[b, l] == v },  B=2048, L=200, V=50000, out f32.
//
// Roofline: 409.6 MB of output writes dominate (~17.6 us at 23.3 TB/s HBM).
// The 1.6 MB index read + 409,600 L2 atomics are noise. Zero FLOPs -> the
// CDNA5 path that matters is the store stream, not the matrix pipes.
//
// Phase 1: zero-fill d_out, 4x global_store_b128 th:TH_STORE_NT per thread
//          (64 B/thread, coalesced; NT since 409.6 MB > 192 MB L2).
// Phase 2: one thread per token, non-returning global_atomic_add_f32.
// Deterministic: all adds are +1.0f, per-slot counts <= 200 (exact in f32).

#define VOCAB 50000

typedef float __attribute__((ext_vector_type(4))) floatx4;

__global__ void __launch_bounds__(256) mh_zero_kernel(floatx4* __restrict__ out,
                                                      long long n4) {
    // Block tile: 1024 float4 = 16 KB. Lane j stores elements
    // {j, j+256, j+512, j+768} of the tile -> consecutive lanes hit
    // consecutive 16 B, every store is a fully coalesced 512 B wave op.
    long long base = (long long)blockIdx.x * 1024 + threadIdx.x;
    floatx4 z = {0.0f, 0.0f, 0.0f, 0.0f};
    if (base + 768 < n4) {
        // Fast path (always taken for the reference shape: 25.6e6 % 1024 == 0):
        // four unconditional b128 NT stores with immediate offsets.
        __builtin_nontemporal_store(z, out + base);
        __builtin_nontemporal_store(z, out + base + 256);
        __builtin_nontemporal_store(z, out + base + 512);
        __builtin_nontemporal_store(z, out + base + 768);
    } else {
        #pragma unroll
        for (int j = 0; j < 4; ++j) {
            long long i = base + (long long)j * 256;
            if (i < n4) __builtin_nontemporal_store(z, out + i);
        }
    }
}

__global__ void __launch_bounds__(256) mh_scatter_kernel(const int* __restrict__ x,
                                                         float* __restrict__ out,
                                                         int total, int L) {
    int i = blockIdx.x * blockDim.x + threadIdx.x;
    if (i >= total) return;
    int b   = i / L;
    int tok = __builtin_nontemporal_load(x + i);   // one-shot streamed index read
    // Defensive clamp (reference guarantees 0 <= tok < VOCAB).
    if (tok < 0) tok = 0;
    if (tok >= VOCAB) tok = VOCAB - 1;
    // Non-returning atomic -> global_atomic_add_f32 (no VGPR writeback),
    // resolved in the L2 atomic units.
    atomicAdd(out + (size_t)b * VOCAB + tok, 1.0f);
}

extern "C" void kernel_launch(void* const* d_in, const int* in_sizes, int n_in,
                              void* d_out, int out_size, void* d_ws, size_t ws_size,
                              hipStream_t stream) {
    const int* x = (const int*)d_in[0];
    float*   out = (float*)d_out;

    const int total = in_sizes[0];      // B * L = 409,600
    const int L     = 200;              // fixed by the reference

    // Phase 1: zero-fill. n4 = 25,600,000 float4; 1024 float4 per block.
    long long n4 = (long long)out_size / 4;
    long long zblocks = (n4 + 1023) / 1024;        // 25,000 blocks x 8 wave32
    mh_zero_kernel<<<(dim3)(unsigned)zblocks, 256, 0, stream>>>((floatx4*)out, n4);

    // Phase 2: scatter-add counts. Stream order guarantees phase 1 completed.
    int sblocks = (total + 255) / 256;             // 1,600 blocks
    mh_scatter_kernel<<<sblocks, 256, 0, stream>>>(x, out, total, L);
}